// approxGNN_9586367004883
// MI455X (gfx1250) — compile-verified
//
#include <hip/hip_runtime.h>
#include <hip/hip_bf16.h>

typedef __attribute__((ext_vector_type(2))) float v2f;
typedef __attribute__((ext_vector_type(8))) float v8f;

#define FEAT 128
#define KSHARP 1000.0f
#define EPSSH  5.0f
#define SMAX   8192   // LDS tile of K*X values (32 KB)

// ---------------------------------------------------------------- fast tanh
__device__ __forceinline__ float fast_tanh(float z) {
#if __has_builtin(__builtin_amdgcn_tanhf)
    return __builtin_amdgcn_tanhf(z);          // v_tanh_f32 (CDNA5 transcendental)
#else
    // tanh(z) = 1 - 2/(exp(2z)+1); exp overflow/underflow saturate to +/-1 correctly
    float e = __expf(2.0f * z);                // v_exp_f32 path
    return 1.0f - 2.0f * __builtin_amdgcn_rcpf(e + 1.0f);
#endif
}

// ------------------------------------------------- 1) aggx = x  (copy init)
__global__ void init_aggx_kernel(const float* __restrict__ x,
                                 float* __restrict__ aggx, int total4) {
    int tid = blockIdx.x * blockDim.x + threadIdx.x;
    if (tid < total4) {
        ((float4*)aggx)[tid] = ((const float4*)x)[tid];
    }
}

// ---------------------------------- 2) aggx[dst] += x[src]  (edge scatter)
__global__ void edge_agg_kernel(const float* __restrict__ x,
                                const int* __restrict__ ei,   // [2,E] (int32)
                                float* __restrict__ aggx, int E) {
    int tid = blockIdx.x * blockDim.x + threadIdx.x;
    int total = E * (FEAT / 4);
    if (tid >= total) return;
    int e = tid >> 5;            // FEAT/4 == 32 chunks per edge
    int c = (tid & 31) << 2;     // feature offset (multiple of 4)
    int s = ei[e];
    int d = ei[E + e];
    float4 v = *(const float4*)(x + (size_t)s * FEAT + c);
    float* dst = aggx + (size_t)d * FEAT + c;
    atomicAdd(dst + 0, v.x);
    atomicAdd(dst + 1, v.y);
    atomicAdd(dst + 2, v.z);
    atomicAdd(dst + 3, v.w);
}

// ------- 3) fused MLP via fp32 WMMA: X = relu(aggx@W1 + b1) @ W2, per node
// One wave (32 lanes) computes 16 node-rows. 8 accumulator tiles span the
// 128 output columns; K-loop of 32 steps of V_WMMA_F32_16X16X4_F32.
// Tail rows are handled by CLAMPING the A-row index (always-valid loads, no
// exec-mask divergence in the K-loop); garbage in a clamped A-row only
// affects that same output row, whose store is guarded below.
__global__ void mlp_wmma_kernel(const float* __restrict__ aggx, // [N,128]
                                const float* __restrict__ W1,   // [128,128] row-major [k][n]
                                const float* __restrict__ b1,   // [128]
                                const float* __restrict__ W2,   // [128]
                                float* __restrict__ X, int n) {
    const int lane   = threadIdx.x & 31;
    const int wave   = threadIdx.x >> 5;
    const int m0     = (blockIdx.x * (blockDim.x >> 5) + wave) * 16;
    if (m0 >= n) return;                      // wave-uniform exit

    const int  lhalf = lane >> 4;             // 0: lanes 0-15, 1: lanes 16-31
    const int  l16   = lane & 15;
    const int  rowA  = min(m0 + l16, n - 1);  // clamped A-matrix row for this lane
    const float* arow = aggx + (size_t)rowA * FEAT;

    v8f c[8];
    #pragma unroll
    for (int t = 0; t < 8; ++t)
        #pragma unroll
        for (int q = 0; q < 8; ++q) c[t][q] = 0.0f;

    for (int k = 0; k < FEAT; k += 4) {
        const int ka = k + (lhalf << 1);      // lanes 16-31 carry K+2,K+3
        v2f a = *(const v2f*)(arow + ka);     // one global_load_b64 for the A frag
        #pragma unroll
        for (int t = 0; t < 8; ++t) {
            const int ncol = (t << 4) + l16;
            v2f b;
            b.x = W1[(size_t)ka * FEAT + ncol];
            b.y = W1[(size_t)(ka + 1) * FEAT + ncol];
            c[t] = __builtin_amdgcn_wmma_f32_16x16x4_f32(
                       false, a, false, b, (short)0, c[t], false, false);
        }
    }

    // bias + relu + dot with W2, still in fragment layout:
    // accumulator VGPR r, lanes 0-15 -> row r, lanes 16-31 -> row r+8; col = t*16 + l16
    float xacc[8];
    #pragma unroll
    for (int r = 0; r < 8; ++r) xacc[r] = 0.0f;
    #pragma unroll
    for (int t = 0; t < 8; ++t) {
        const int ncol = (t << 4) + l16;
        const float bias = b1[ncol];
        const float w2   = W2[ncol];
        #pragma unroll
        for (int r = 0; r < 8; ++r) {
            float h = fmaxf(c[t][r] + bias, 0.0f);
            xacc[r] = fmaf(h, w2, xacc[r]);
        }
    }
    // reduce the 16 lanes of each half (rows r and r+8 independently)
    #pragma unroll
    for (int r = 0; r < 8; ++r) {
        float v = xacc[r];
        v += __shfl_xor(v, 1, 16);
        v += __shfl_xor(v, 2, 16);
        v += __shfl_xor(v, 4, 16);
        v += __shfl_xor(v, 8, 16);
        xacc[r] = v;
    }
    if (l16 == 0) {
        #pragma unroll
        for (int r = 0; r < 8; ++r) {
            const int row = m0 + r + (lhalf << 3);
            if (row < n) X[row] = xacc[r];
        }
    }
}

// ------------- 4) out[i] = sum_j tanh(K*(X[i]-X[j]) - EPS)   (O(N^2) path)
__global__ void pairwise_kernel(const float* __restrict__ X,
                                float* __restrict__ out, int n) {
    __shared__ float kx[SMAX];
    const int tid = threadIdx.x;
    const int i   = blockIdx.x * blockDim.x + tid;
    const float ai = (i < n) ? fmaf(KSHARP, X[i], -EPSSH) : 0.0f;
    float s = 0.0f;

    for (int base = 0; base < n; base += SMAX) {
        const int cnt = min(SMAX, n - base);
        for (int j = tid; j < cnt; j += blockDim.x)
            kx[j] = KSHARP * X[base + j];
        __syncthreads();

        const int cnt4 = cnt & ~3;
        for (int j = 0; j < cnt4; j += 4) {
            float4 k4 = *(const float4*)&kx[j];
            s += fast_tanh(ai - k4.x);
            s += fast_tanh(ai - k4.y);
            s += fast_tanh(ai - k4.z);
            s += fast_tanh(ai - k4.w);
        }
        for (int j = cnt4; j < cnt; ++j)
            s += fast_tanh(ai - kx[j]);
        __syncthreads();
    }
    if (i < n) out[i] = s;
}

// -----------------------------------------------------------------------
extern "C" void kernel_launch(void* const* d_in, const int* in_sizes, int n_in,
                              void* d_out, int out_size, void* d_ws, size_t ws_size,
                              hipStream_t stream) {
    const float* x  = (const float*)d_in[0];
    const int*   ei = (const int*)d_in[1];
    const float* W1 = (const float*)d_in[2];
    const float* b1 = (const float*)d_in[3];
    const float* W2 = (const float*)d_in[4];
    float* out = (float*)d_out;

    const int N = in_sizes[0] / FEAT;
    const int E = in_sizes[1] / 2;

    float* aggx = (float*)d_ws;                       // [N,128]
    float* X    = aggx + (size_t)N * FEAT;            // [N]

    // 1) aggx = x  (folds the "x + agg" add into the scatter target)
    const int total4 = N * (FEAT / 4);
    init_aggx_kernel<<<(total4 + 255) / 256, 256, 0, stream>>>(x, aggx, total4);

    // 2) aggx[dst] += x[src]
    const int tE = E * (FEAT / 4);
    edge_agg_kernel<<<(tE + 255) / 256, 256, 0, stream>>>(x, ei, aggx, E);

    // 3) X = relu(aggx@W1 + b1) @ W2   (fp32 WMMA, 4 waves / 64 rows per block)
    const int rowTiles = (N + 15) / 16;
    const int mlpBlocks = (rowTiles + 3) / 4;
    mlp_wmma_kernel<<<mlpBlocks, 128, 0, stream>>>(aggx, W1, b1, W2, X, N);

    // 4) O(N^2) tanh row-sum
    pairwise_kernel<<<(N + 255) / 256, 256, 0, stream>>>(X, out, N);
}